// FirstGCN_9749575762778
// MI455X (gfx1250) — compile-verified
//
#include <hip/hip_runtime.h>

typedef __attribute__((ext_vector_type(2))) float v2f;
typedef __attribute__((ext_vector_type(8))) float v8f;

#define NNODES 100000
#define DFEAT  128
#define NEDGE  600000

// ---------------------------------------------------------------- utilities
__global__ void zero_kernel(float* __restrict__ p, long n) {
  long i = (long)blockIdx.x * blockDim.x + threadIdx.x;
  long stride = (long)gridDim.x * blockDim.x;
  for (; i < n; i += stride) p[i] = 0.0f;
}

__global__ void relu_kernel(float* __restrict__ p, long n) {
  long i = (long)blockIdx.x * blockDim.x + threadIdx.x;
  long stride = (long)gridDim.x * blockDim.x;
  for (; i < n; i += stride) p[i] = fmaxf(p[i], 0.0f);
}

// -------------------------------------------------------- degree + norms
__global__ void degree_kernel(const int* __restrict__ s, const int* __restrict__ d,
                              float* __restrict__ degs, float* __restrict__ degd, int e) {
  int i = blockIdx.x * blockDim.x + threadIdx.x;
  if (i < e) {
    atomicAdd(&degs[s[i]], 1.0f);
    atomicAdd(&degd[d[i]], 1.0f);
  }
}

__global__ void norm_kernel(float* __restrict__ deg, int n) {
  int i = blockIdx.x * blockDim.x + threadIdx.x;
  if (i < n) deg[i] = rsqrtf(fmaxf(deg[i], 1.0f));
}

// ------------------------------------------- edge gather + scatter-add (L2)
// One wave32 per edge: each lane moves 4 floats (float4) of the 128-float row.
__global__ void scatter_kernel(const float* __restrict__ x,
                               const int* __restrict__ s,
                               const int* __restrict__ d,
                               const float* __restrict__ norms,
                               float* __restrict__ agg, int e) {
  int gtid = blockIdx.x * blockDim.x + threadIdx.x;
  int wave = gtid >> 5;
  int lane = gtid & 31;
  if (wave >= e) return;

  // prefetch a future edge's source row into cache (global_prefetch_b8)
  int pe = wave + 64;
  if (pe < e) {
    __builtin_prefetch(x + (size_t)s[pe] * DFEAT + lane * 4, 0, 3);
  }

  int sv = s[wave];
  int dv = d[wave];
  float ns = norms[sv];                                  // broadcast per wave
  const float4* xs = (const float4*)(x + (size_t)sv * DFEAT);
  float4 v = xs[lane];                                   // 32 lanes x 16B = 512B row
  float* ag = agg + (size_t)dv * DFEAT + lane * 4;
  atomicAdd(ag + 0, v.x * ns);                           // global_atomic_add_f32
  atomicAdd(ag + 1, v.y * ns);                           // agg fits in 192MB L2
  atomicAdd(ag + 2, v.z * ns);
  atomicAdd(ag + 3, v.w * ns);
}

// -------------------------------------- projection: (agg * norm_d) @ W + b
// out[dt] += h, fused with norm_d scaling on A-fragment loads.
// WMMA f32 16x16x4: one wave per (16-row M-tile, 16-col N-tile).
// Block = 256 threads = 8 waves; wave w owns N-tile w (D=128 = 8 tiles).
// Grid.x = NNODES/16 = 6250 (exact; no predication -> EXEC all-1s for WMMA).
__global__ __launch_bounds__(256) void gemm_kernel(const float* __restrict__ agg,
                                                   const float* __restrict__ normd,
                                                   const float* __restrict__ W,
                                                   const float* __restrict__ bias,
                                                   float* __restrict__ out) {
  const int lane  = threadIdx.x & 31;
  const int wv    = threadIdx.x >> 5;          // N-tile index 0..7
  const int mbase = blockIdx.x * 16;
  const int l15   = lane & 15;
  const int mrow  = mbase + l15;               // A-fragment row for this lane
  const float nd  = normd[mrow];               // fuse norm_d into A load
  const int koff  = (lane < 16) ? 0 : 2;       // lanes 16-31 hold K+2,K+3
  const int col   = wv * 16 + l15;             // B/C/D column for this lane

  const float* arow = agg + (size_t)mrow * DFEAT + koff;

  v8f acc = {};
#pragma unroll 4
  for (int k0 = 0; k0 < DFEAT; k0 += 4) {
    v2f a, bf;
    // A 16x4 fragment: VGPR0 = K=k0(+2), VGPR1 = K=k0+1(+3)
    a.x = arow[k0]     * nd;
    a.y = arow[k0 + 1] * nd;
    // B 4x16 fragment: lane = N, VGPR0 = K=k0(+2), VGPR1 = K=k0+1(+3)
    bf.x = W[(size_t)(k0 + koff)     * DFEAT + col];
    bf.y = W[(size_t)(k0 + koff + 1) * DFEAT + col];
    acc = __builtin_amdgcn_wmma_f32_16x16x4_f32(
        /*neg_a=*/false, a, /*neg_b=*/false, bf,
        /*c_mod=*/(short)0, acc, /*reuse_a=*/false, /*reuse_b=*/false);
  }

  // C/D layout: VGPR i -> M = mbase + i (+8 for lanes 16-31), N = col
  const float bv = bias[col];
  const int m2 = mbase + ((lane < 16) ? 0 : 8);
#pragma unroll
  for (int i = 0; i < 8; ++i) {
    out[(size_t)(m2 + i) * DFEAT + col] += acc[i] + bv;   // serialized per relation
  }
}

// ---------------------------------------------------------------- driver
static const int REL_SRC[9] = {0, 0, 0, 1, 1, 2, 2, 2, 1};
static const int REL_DST[9] = {0, 1, 2, 2, 0, 0, 1, 2, 1};

extern "C" void kernel_launch(void* const* d_in, const int* in_sizes, int n_in,
                              void* d_out, int out_size, void* d_ws, size_t ws_size,
                              hipStream_t stream) {
  const float* x   = (const float*)d_in[0];   // [3, N, D]
  const float* W   = (const float*)d_in[1];   // [9, D, D]
  const float* b   = (const float*)d_in[2];   // [9, D]
  const int*   src = (const int*)d_in[3];     // [9, E]
  const int*   dst = (const int*)d_in[4];     // [9, E]
  float* out = (float*)d_out;                 // [3, N, D]

  float* ws    = (float*)d_ws;
  float* deg_s = ws;                          // [N]  (becomes norm_s)
  float* deg_d = ws + NNODES;                 // [N]  (becomes norm_d)
  float* agg   = ws + 2 * NNODES;             // [N, D] — 51 MB, L2-resident

  const long outn   = 3L * NNODES * DFEAT;
  const long clearn = 2L * NNODES + (long)NNODES * DFEAT;

  zero_kernel<<<2048, 256, 0, stream>>>(out, outn);

  for (int r = 0; r < 9; ++r) {
    const int st = REL_SRC[r];
    const int dt = REL_DST[r];
    const int* s_r = src + (size_t)r * NEDGE;
    const int* d_r = dst + (size_t)r * NEDGE;

    zero_kernel<<<2048, 256, 0, stream>>>(ws, clearn);
    degree_kernel<<<(NEDGE + 255) / 256, 256, 0, stream>>>(s_r, d_r, deg_s, deg_d, NEDGE);
    norm_kernel<<<(2 * NNODES + 255) / 256, 256, 0, stream>>>(ws, 2 * NNODES);
    scatter_kernel<<<(NEDGE * 32 + 255) / 256, 256, 0, stream>>>(
        x + (size_t)st * NNODES * DFEAT, s_r, d_r, deg_s, agg, NEDGE);
    gemm_kernel<<<NNODES / 16, 256, 0, stream>>>(
        agg, deg_d, W + (size_t)r * DFEAT * DFEAT, b + (size_t)r * DFEAT,
        out + (size_t)dt * NNODES * DFEAT);
  }

  relu_kernel<<<2048, 256, 0, stream>>>(out, outn);
}